// ARMA_7103875907623
// MI455X (gfx1250) — compile-verified
//
#include <hip/hip_runtime.h>
#include <hip/hip_bf16.h>

typedef float v2f __attribute__((ext_vector_type(2)));
typedef float v8f __attribute__((ext_vector_type(8)));

#define FIN 75
#define FINP 76   // padded to multiple of 4
#define HID 16
#define FOUT 64
#define KST 3

// ---------------- padding / elementwise / graph kernels ----------------

__global__ void zero_kernel(float* __restrict__ p, long n) {
  long i = (long)blockIdx.x * blockDim.x + threadIdx.x;
  long stride = (long)gridDim.x * blockDim.x;
  for (; i < n; i += stride) p[i] = 0.0f;
}

// xp[n,f] (Npad x 76) = zero-padded copy of x (nN x 75)
__global__ void pad_x_kernel(const float* __restrict__ x, float* __restrict__ xp,
                             int nN, int Npad) {
  long i = (long)blockIdx.x * blockDim.x + threadIdx.x;
  long total = (long)Npad * FINP;
  if (i >= total) return;
  int n = (int)(i / FINP);
  int f = (int)(i % FINP);
  xp[i] = (n < nN && f < FIN) ? x[(long)n * FIN + f] : 0.0f;
}

// w1p layout: [0..3*76*16) = padded w1_init, [3*76*16..) = padded w1_root
__global__ void pad_w1_kernel(const float* __restrict__ wi, const float* __restrict__ wr,
                              float* __restrict__ w1p) {
  int i = blockIdx.x * blockDim.x + threadIdx.x;
  const int per = KST * FINP * HID;
  if (i >= per) return;
  int k = i / (FINP * HID);
  int rem = i % (FINP * HID);
  int f = rem / HID;
  int h = rem % HID;
  float vi = (f < FIN) ? wi[((long)k * FIN + f) * HID + h] : 0.0f;
  float vr = (f < FIN) ? wr[((long)k * FIN + f) * HID + h] : 0.0f;
  w1p[i] = vi;
  w1p[per + i] = vr;
}

__global__ void deg_kernel(const int* __restrict__ col, float* __restrict__ deg, int nE) {
  int e = blockIdx.x * blockDim.x + threadIdx.x;
  if (e < nE) atomicAdd(&deg[col[e]], 1.0f);
}

__global__ void dinv_kernel(float* __restrict__ deg, int nN) {
  int i = blockIdx.x * blockDim.x + threadIdx.x;
  if (i < nN) {
    float d = deg[i];
    deg[i] = (d > 0.0f) ? rsqrtf(d) : 0.0f;
  }
}

__global__ void norm_kernel(const int* __restrict__ row, const int* __restrict__ col,
                            const float* __restrict__ dinv, float* __restrict__ norm, int nE) {
  int e = blockIdx.x * blockDim.x + threadIdx.x;
  if (e < nE) norm[e] = dinv[row[e]] * dinv[col[e]];
}

// scatter-add propagation, layer1 ([KST,Npad,HID]); 4 lanes/edge, float4 each
__global__ void prop16_kernel(const float* __restrict__ src, float* __restrict__ dst,
                              const float* __restrict__ norm, const int* __restrict__ row,
                              const int* __restrict__ col, int Npad, int nE) {
  long t = (long)blockIdx.x * blockDim.x + threadIdx.x;
  long total = (long)nE * 4;
  if (t >= total) return;
  int e = (int)(t >> 2);
  int h4 = (int)(t & 3) * 4;
  float nm = norm[e];
  int r = row[e], c = col[e];
#pragma unroll
  for (int k = 0; k < KST; ++k) {
    const float4 v = *(const float4*)(src + ((long)k * Npad + r) * HID + h4);
    float* d = dst + ((long)k * Npad + c) * HID + h4;
    atomicAdd(d + 0, v.x * nm);
    atomicAdd(d + 1, v.y * nm);
    atomicAdd(d + 2, v.z * nm);
    atomicAdd(d + 3, v.w * nm);
  }
}

// scatter-add propagation, layer2 per-stack ([Npad,FOUT]); 16 lanes/edge, float4 each
__global__ void prop64_kernel(const float* __restrict__ src, float* __restrict__ dst,
                              const float* __restrict__ norm, const int* __restrict__ row,
                              const int* __restrict__ col, int nE) {
  long t = (long)blockIdx.x * blockDim.x + threadIdx.x;
  long total = (long)nE * 16;
  if (t >= total) return;
  int e = (int)(t >> 4);
  int o4 = (int)(t & 15) * 4;
  float nm = norm[e];
  const float4 v = *(const float4*)(src + (long)row[e] * FOUT + o4);
  float* d = dst + (long)col[e] * FOUT + o4;
  atomicAdd(d + 0, v.x * nm);
  atomicAdd(d + 1, v.y * nm);
  atomicAdd(d + 2, v.z * nm);
  atomicAdd(d + 3, v.w * nm);
}

__global__ void combine_relu_kernel(const float* __restrict__ a, const float* __restrict__ b,
                                    float* __restrict__ out, long n) {
  long i = (long)blockIdx.x * blockDim.x + threadIdx.x;
  if (i < n) out[i] = fmaxf(a[i] + b[i], 0.0f);
}

__global__ void combine_kernel(const float* __restrict__ a, const float* __restrict__ b,
                               float* __restrict__ out, long n) {
  long i = (long)blockIdx.x * blockDim.x + threadIdx.x;
  if (i < n) out[i] = a[i] + b[i];
}

// h1[n,h] = relu(mean_k state[k,n,h]) over padded rows (pad-row garbage never leaks)
__global__ void mean16_relu_kernel(const float* __restrict__ state, float* __restrict__ h1,
                                   int Npad) {
  long i = (long)blockIdx.x * blockDim.x + threadIdx.x;
  long n = (long)Npad * HID;
  if (i < n) {
    float s = state[i] + state[n + i] + state[2 * n + i];
    h1[i] = fmaxf(s * (1.0f / 3.0f), 0.0f);
  }
}

__global__ void out_init_kernel(float* __restrict__ out, const float* __restrict__ bg, int G) {
  int g = blockIdx.x * blockDim.x + threadIdx.x;
  if (g < G) out[g] = bg[0];
}

// out[batch[n]] += (h2[n,:] . wg) / 3   (per-stack contribution of the mean)
__global__ void head_kernel(const float* __restrict__ h2, const float* __restrict__ wg,
                            const int* __restrict__ batch, float* __restrict__ out, int nN) {
  int n = blockIdx.x * blockDim.x + threadIdx.x;
  if (n >= nN) return;
  const float4* hp = (const float4*)(h2 + (long)n * FOUT);
  const float4* wp = (const float4*)wg;
  float dot = 0.0f;
#pragma unroll
  for (int i = 0; i < FOUT / 4; ++i) {
    float4 a = hp[i], w = wp[i];
    dot += a.x * w.x + a.y * w.y + a.z * w.z + a.w * w.w;
  }
  atomicAdd(&out[batch[n]], dot * (1.0f / 3.0f));
}

// ---------------- WMMA GEMM kernels (f32 16x16x4), fully unguarded ----------------
// A 16x4 (MxK) f32: VGPR v, lane-half s: K = 2*s + v ; M = lane&15
// B 4x16  (KxN) f32: VGPR v, lane-half s: row K = 2*s + v, col = lane&15
// D 16x16 f32: VGPR v, lane l: row = v + 8*(l>>4), col = l&15

// state[k,n,h] = xp[n,:]@w1p_init[k]; xroot[k,n,h] = xp[n,:]@w1p_root[k] + bias[k,h]
__global__ void __launch_bounds__(32) l1_in_gemm(
    const float* __restrict__ xp, const float* __restrict__ w1p,
    const float* __restrict__ bias, float* __restrict__ state,
    float* __restrict__ xroot, int Npad) {
  const int n0 = blockIdx.x * 16;
  const int k = blockIdx.y;
  const int lane = threadIdx.x;
  const int half = lane >> 4;
  const int l16 = lane & 15;
  const float* xr = xp + (long)(n0 + l16) * FINP;
  const float* wi = w1p + (long)k * FINP * HID;
  const float* wr = w1p + (long)(KST + k) * FINP * HID;
  v8f acc_i = {};
  v8f acc_r = {};
  for (int c = 0; c < FINP / 4; ++c) {
    int f0 = c * 4 + half * 2;
    v2f a = *(const v2f*)(xr + f0);
    v2f bi, br;
    bi.x = wi[(f0 + 0) * HID + l16];
    bi.y = wi[(f0 + 1) * HID + l16];
    br.x = wr[(f0 + 0) * HID + l16];
    br.y = wr[(f0 + 1) * HID + l16];
    acc_i = __builtin_amdgcn_wmma_f32_16x16x4_f32(false, a, false, bi, (short)0, acc_i, false, false);
    acc_r = __builtin_amdgcn_wmma_f32_16x16x4_f32(false, a, false, br, (short)0, acc_r, false, false);
  }
  float bv = bias[k * HID + l16];
#pragma unroll
  for (int v = 0; v < 8; ++v) {
    int nn = n0 + v + 8 * half;
    state[((long)k * Npad + nn) * HID + l16] = acc_i[v];
    xroot[((long)k * Npad + nn) * HID + l16] = acc_r[v] + bv;
  }
}

// in-place: state[k,n,:] = state[k,n,:] @ w[k]   (16x16)
__global__ void __launch_bounds__(32) rec16_gemm(float* __restrict__ state,
                                                 const float* __restrict__ w, int Npad) {
  const int n0 = blockIdx.x * 16;
  const int k = blockIdx.y;
  const int lane = threadIdx.x;
  const int half = lane >> 4;
  const int l16 = lane & 15;
  const float* srow = state + ((long)k * Npad + n0 + l16) * HID;
  const float* wk = w + k * HID * HID;
  v8f acc = {};
#pragma unroll
  for (int c = 0; c < 4; ++c) {
    int f0 = c * 4 + half * 2;
    v2f a = *(const v2f*)(srow + f0);
    v2f b;
    b.x = wk[(f0 + 0) * HID + l16];
    b.y = wk[(f0 + 1) * HID + l16];
    acc = __builtin_amdgcn_wmma_f32_16x16x4_f32(false, a, false, b, (short)0, acc, false, false);
  }
#pragma unroll
  for (int v = 0; v < 8; ++v) {
    int nn = n0 + v + 8 * half;
    state[((long)k * Npad + nn) * HID + l16] = acc[v];
  }
}

// state[n,o]=h1[n,:]@w2_init[k]; xroot[n,o]=h1[n,:]@w2_root[k]+bias[k,o]   (16 -> 64)
__global__ void __launch_bounds__(32) l2_in_gemm(
    const float* __restrict__ h1, const float* __restrict__ w_init,
    const float* __restrict__ w_root, const float* __restrict__ bias,
    float* __restrict__ state, float* __restrict__ xroot, int Npad, int k) {
  const int n0 = blockIdx.x * 16;
  const int o0 = blockIdx.y * 16;
  const int lane = threadIdx.x;
  const int half = lane >> 4;
  const int l16 = lane & 15;
  const float* hrow = h1 + (long)(n0 + l16) * HID;
  const float* wi = w_init + (long)k * HID * FOUT;
  const float* wr = w_root + (long)k * HID * FOUT;
  v8f acc_i = {};
  v8f acc_r = {};
#pragma unroll
  for (int c = 0; c < 4; ++c) {
    int f0 = c * 4 + half * 2;
    v2f a = *(const v2f*)(hrow + f0);
    v2f bi, br;
    bi.x = wi[(f0 + 0) * FOUT + o0 + l16];
    bi.y = wi[(f0 + 1) * FOUT + o0 + l16];
    br.x = wr[(f0 + 0) * FOUT + o0 + l16];
    br.y = wr[(f0 + 1) * FOUT + o0 + l16];
    acc_i = __builtin_amdgcn_wmma_f32_16x16x4_f32(false, a, false, bi, (short)0, acc_i, false, false);
    acc_r = __builtin_amdgcn_wmma_f32_16x16x4_f32(false, a, false, br, (short)0, acc_r, false, false);
  }
  float bv = bias[k * FOUT + o0 + l16];
#pragma unroll
  for (int v = 0; v < 8; ++v) {
    int nn = n0 + v + 8 * half;
    state[(long)nn * FOUT + o0 + l16] = acc_i[v];
    xroot[(long)nn * FOUT + o0 + l16] = acc_r[v] + bv;
  }
}

// in-place: state[n,:] = state[n,:] @ w2_w[k]  (64x64); wave owns 16 rows, A loaded first
__global__ void __launch_bounds__(32) rec64_gemm(float* __restrict__ state,
                                                 const float* __restrict__ w, int Npad, int k) {
  const int n0 = blockIdx.x * 16;
  const int lane = threadIdx.x;
  const int half = lane >> 4;
  const int l16 = lane & 15;
  const float* srow = state + (long)(n0 + l16) * FOUT;
  const float* wk = w + (long)k * FOUT * FOUT;
  v2f a[16];
#pragma unroll
  for (int c = 0; c < 16; ++c) {
    int f0 = c * 4 + half * 2;
    a[c] = *(const v2f*)(srow + f0);
  }
#pragma unroll
  for (int ct = 0; ct < 4; ++ct) {
    int o0 = ct * 16;
    v8f acc = {};
#pragma unroll
    for (int c = 0; c < 16; ++c) {
      int f0 = c * 4 + half * 2;
      v2f b;
      b.x = wk[(f0 + 0) * FOUT + o0 + l16];
      b.y = wk[(f0 + 1) * FOUT + o0 + l16];
      acc = __builtin_amdgcn_wmma_f32_16x16x4_f32(false, a[c], false, b, (short)0, acc, false, false);
    }
#pragma unroll
    for (int v = 0; v < 8; ++v) {
      int nn = n0 + v + 8 * half;
      state[(long)nn * FOUT + o0 + l16] = acc[v];
    }
  }
}

// ---------------- host orchestration ----------------

static inline unsigned gs256(long n) { return (unsigned)((n + 255) / 256); }

extern "C" void kernel_launch(void* const* d_in, const int* in_sizes, int n_in,
                              void* d_out, int out_size, void* d_ws, size_t ws_size,
                              hipStream_t stream) {
  const float* x       = (const float*)d_in[0];
  const int*   ei      = (const int*)d_in[1];
  const int*   batch   = (const int*)d_in[2];
  const float* w1_init = (const float*)d_in[3];
  const float* w1_w    = (const float*)d_in[4];
  const float* w1_root = (const float*)d_in[5];
  const float* w1_bias = (const float*)d_in[6];
  const float* w2_init = (const float*)d_in[7];
  const float* w2_w    = (const float*)d_in[8];
  const float* w2_root = (const float*)d_in[9];
  const float* w2_bias = (const float*)d_in[10];
  const float* wg      = (const float*)d_in[11];
  const float* bg      = (const float*)d_in[12];
  float* out = (float*)d_out;
  (void)n_in; (void)ws_size;

  const int nN = in_sizes[0] / FIN;
  const int nE = in_sizes[1] / 2;
  const int G  = out_size;
  const int tiles = (nN + 15) / 16;
  const int Npad = tiles * 16;   // N=100000 -> Npad == N
  const int* row = ei;
  const int* col = ei + nE;

  // ws layout (floats, all region sizes multiples of 4 -> 16B aligned)
  float* ws = (float*)d_ws;
  size_t off = 0;
  float* norm = ws + off; off += ((size_t)nE + 3) & ~3ull;
  float* dinv = ws + off; off += ((size_t)nN + 3) & ~3ull;
  float* xp   = ws + off; off += (size_t)Npad * FINP;
  float* w1p  = ws + off; off += (size_t)2 * KST * FINP * HID;
  float* h1   = ws + off; off += (size_t)Npad * HID;
  const size_t bufElems = (size_t)Npad * FOUT;  // >= 3*Npad*HID
  float* bufA = ws + off; off += bufElems;      // state
  float* bufB = ws + off; off += bufElems;      // scatter accumulator
  float* bufC = ws + off; off += bufElems;      // root + bias

  const long n1 = (long)KST * Npad * HID;
  const long n2 = (long)Npad * FOUT;

  // padding + gcn_norm
  pad_x_kernel<<<gs256((long)Npad * FINP), 256, 0, stream>>>(x, xp, nN, Npad);
  pad_w1_kernel<<<gs256(KST * FINP * HID), 256, 0, stream>>>(w1_init, w1_root, w1p);
  zero_kernel<<<gs256(nN), 256, 0, stream>>>(dinv, nN);
  deg_kernel<<<gs256(nE), 256, 0, stream>>>(col, dinv, nE);
  dinv_kernel<<<gs256(nN), 256, 0, stream>>>(dinv, nN);
  norm_kernel<<<gs256(nE), 256, 0, stream>>>(row, col, dinv, norm, nE);

  // ---- ARMA layer 1 (75 -> 16, K=3, T=2, relu) ----
  l1_in_gemm<<<dim3(tiles, KST), 32, 0, stream>>>(xp, w1p, w1_bias, bufA, bufC, Npad);
  // t=0
  zero_kernel<<<gs256(n1), 256, 0, stream>>>(bufB, n1);
  prop16_kernel<<<gs256((long)nE * 4), 256, 0, stream>>>(bufA, bufB, norm, row, col, Npad, nE);
  combine_relu_kernel<<<gs256(n1), 256, 0, stream>>>(bufB, bufC, bufA, n1);
  // t=1
  rec16_gemm<<<dim3(tiles, KST), 32, 0, stream>>>(bufA, w1_w, Npad);
  zero_kernel<<<gs256(n1), 256, 0, stream>>>(bufB, n1);
  prop16_kernel<<<gs256((long)nE * 4), 256, 0, stream>>>(bufA, bufB, norm, row, col, Npad, nE);
  combine_relu_kernel<<<gs256(n1), 256, 0, stream>>>(bufB, bufC, bufA, n1);
  // mean over stacks + outer relu
  mean16_relu_kernel<<<gs256((long)Npad * HID), 256, 0, stream>>>(bufA, h1, Npad);

  // ---- ARMA layer 2 (16 -> 64, K=3, T=2, no act), per stack; fused pool+head ----
  out_init_kernel<<<gs256(G), 256, 0, stream>>>(out, bg, G);
  for (int k = 0; k < KST; ++k) {
    l2_in_gemm<<<dim3(tiles, 4), 32, 0, stream>>>(h1, w2_init, w2_root, w2_bias, bufA, bufC, Npad, k);
    // t=0
    zero_kernel<<<gs256(n2), 256, 0, stream>>>(bufB, n2);
    prop64_kernel<<<gs256((long)nE * 16), 256, 0, stream>>>(bufA, bufB, norm, row, col, nE);
    combine_kernel<<<gs256(n2), 256, 0, stream>>>(bufB, bufC, bufA, n2);
    // t=1
    rec64_gemm<<<tiles, 32, 0, stream>>>(bufA, w2_w, Npad, k);
    zero_kernel<<<gs256(n2), 256, 0, stream>>>(bufB, n2);
    prop64_kernel<<<gs256((long)nE * 16), 256, 0, stream>>>(bufA, bufB, norm, row, col, nE);
    combine_kernel<<<gs256(n2), 256, 0, stream>>>(bufB, bufC, bufA, n2);
    // pooled head: out[batch[n]] += (state[n,:] . wg)/3
    head_kernel<<<gs256(nN), 256, 0, stream>>>(bufA, wg, batch, out, nN);
  }
}